// NNUE_38869454029100
// MI455X (gfx1250) — compile-verified
//
#include <hip/hip_runtime.h>

#define K_IN   40960
#define HID    128
#define BATCH  4096
#define KCHUNK 128
#define PITCH  132   // 128 + 4 floats pad -> conflict-free ds_load_b64 B-reads

#if defined(__has_builtin)
# if __has_builtin(__builtin_amdgcn_global_load_async_to_lds_b128)
#  define HAVE_ASYNC_LDS 1
# endif
#endif
#ifndef HAVE_ASYNC_LDS
# define HAVE_ASYNC_LDS 0
#endif

#if HAVE_ASYNC_LDS
# if __has_builtin(__builtin_amdgcn_s_wait_asynccnt)
#  define WAIT_ASYNC(n) __builtin_amdgcn_s_wait_asynccnt(n)
# else
#  define WAIT_ASYNC(n) asm volatile("s_wait_asynccnt " #n ::: "memory")
# endif
#endif

typedef float v2f __attribute__((ext_vector_type(2)));
typedef float v8f __attribute__((ext_vector_type(8)));

// Builtin parameter types (from hipcc diagnostic): non-const pointer to a
// 16-byte int vector; arg0 in the global (AS1, printed "__device__") address
// space, arg1 in LDS (AS3).
typedef int v4i_g __attribute__((vector_size(16)));
typedef __attribute__((address_space(1))) v4i_g* gas_b128_ptr;
typedef __attribute__((address_space(3))) v4i_g* las_b128_ptr;

__global__ void nnue_zero_out(float* __restrict__ out) {
    int i = blockIdx.x * blockDim.x + threadIdx.x;
    if (i < BATCH) out[i] = 0.0f;
}

// Stage a 64-float (16B x 16) strip of the weight chunk for this thread.
// Async path: global -> LDS DMA tracked by ASYNCcnt (no VGPR round trip).
__device__ __forceinline__
void stage_chunk(const float* __restrict__ gsrc, float* ldst)
{
#if HAVE_ASYNC_LDS
    #pragma unroll
    for (int i = 0; i < 16; ++i)
        __builtin_amdgcn_global_load_async_to_lds_b128(
            (gas_b128_ptr)(float*)(gsrc + i * 4),
            (las_b128_ptr)(ldst + i * 4),
            /*offset=*/0, /*cpol=*/0);
#else
    #pragma unroll
    for (int i = 0; i < 16; ++i)
        ((float4*)ldst)[i] = ((const float4*)gsrc)[i];
#endif
}

// One wave = 16 batch rows x all 128 hidden cols (8 f32 WMMA accumulators).
// 8 waves/block share the weight chunk, double-buffered in LDS and prefetched
// one chunk ahead via async global->LDS. FUSED: relu + w2-dot + atomicAdd
// (exactly 2 commutative adds per out element -> deterministic).
// !FUSED: write pre-ReLU partials to workspace for the split-K reduce.
template <bool FUSED>
__global__ __launch_bounds__(256, 2)
void nnue_gemm(const float* __restrict__ x1, const float* __restrict__ x2,
               const float* __restrict__ w_us, const float* __restrict__ w_them,
               const float* __restrict__ w2, float* __restrict__ out,
               float* __restrict__ partial, int KS, int klen)
{
    __shared__ float ldsW[2][HID * PITCH];

    const int half = blockIdx.z;                // 0: (x1,w_us)  1: (x2,w_them)
    const int ks   = blockIdx.y;                // K-split index
    const float* __restrict__ x = half ? x2 : x1;
    const float* __restrict__ w = half ? w_them : w_us;

    const int tid  = threadIdx.x;
    const int wave = tid >> 5;
    const int lane = tid & 31;
    const int l16  = lane & 15;
    const int hi   = lane >> 4;                 // 0 or 1 (half-wave)

    const int m0 = blockIdx.x * 128 + wave * 16;

    v8f acc[8] = {};

    // A operand: per-lane row, float2 at k + 2*hi (16x4 f32 A layout)
    const float* __restrict__ xrow = x + (size_t)(m0 + l16) * K_IN;

    // LDS staging map: thread -> hidden row sn, 64-float K sub-chunk sk0
    const int sn  = tid >> 1;
    const int sk0 = (tid & 1) * 64;
    const float* __restrict__ wsrcBase = w + (size_t)sn * K_IN + sk0;
    float* const ldst0 = &ldsW[0][sn * PITCH + sk0];
    float* const ldst1 = &ldsW[1][sn * PITCH + sk0];

    const int kbeg = ks * klen;
    const int nch  = klen / KCHUNK;

    // prologue: start chunk 0 into buffer 0
    stage_chunk(wsrcBase + kbeg, ldst0);

    for (int c = 0; c < nch; ++c) {
        const int k0 = kbeg + c * KCHUNK;
        const float* __restrict__ cbuf = (c & 1) ? &ldsW[1][0] : &ldsW[0][0];

        // prefetch next chunk into the other buffer (consumed last iteration,
        // all waves are past it thanks to the trailing barrier)
        if (c + 1 < nch)
            stage_chunk(wsrcBase + k0 + KCHUNK, (c & 1) ? ldst0 : ldst1);

#if HAVE_ASYNC_LDS
        // async loads complete in order: <=16 outstanding means the current
        // chunk's 16 ops are done; only the prefetch remains in flight.
        if (c + 1 < nch) WAIT_ASYNC(16);
        else             WAIT_ASYNC(0);
#endif
        __builtin_prefetch(xrow + k0 + KCHUNK, 0, 0);
        __syncthreads();

        #pragma unroll 8
        for (int kk = 0; kk < KCHUNK; kk += 4) {
            v2f a = *(const v2f*)(xrow + k0 + kk + 2 * hi);
            #pragma unroll
            for (int t = 0; t < 8; ++t) {
                // B operand: w^T[k..k+3][n], float2 at lds[n*PITCH + kk + 2*hi]
                v2f b = *(const v2f*)(cbuf + (t * 16 + l16) * PITCH + kk + 2 * hi);
                acc[t] = __builtin_amdgcn_wmma_f32_16x16x4_f32(
                    false, a, false, b, (short)0, acc[t], false, false);
            }
        }
        __syncthreads();
    }

    if (FUSED) {
        // C/D layout: acc[t][j] is row m0 + j + 8*hi, col t*16 + l16
        float w2v[8];
        #pragma unroll
        for (int t = 0; t < 8; ++t) w2v[t] = w2[half * HID + t * 16 + l16];
        #pragma unroll
        for (int j = 0; j < 8; ++j) {
            float s = 0.0f;
            #pragma unroll
            for (int t = 0; t < 8; ++t) {
                float c = acc[t][j];
                c = c > 0.0f ? c : 0.0f;           // ReLU
                s += c * w2v[t];
            }
            // reduce the 16 lanes of each half-wave (fixed-order -> deterministic)
            s += __shfl_xor(s, 1, 32);
            s += __shfl_xor(s, 2, 32);
            s += __shfl_xor(s, 4, 32);
            s += __shfl_xor(s, 8, 32);
            if (l16 == 0) atomicAdd(out + m0 + hi * 8 + j, s);
        }
    } else {
        float* __restrict__ pbase =
            partial + ((size_t)(half * KS + ks) * BATCH) * HID;
        #pragma unroll
        for (int j = 0; j < 8; ++j) {
            float* __restrict__ prow = pbase + (size_t)(m0 + hi * 8 + j) * HID + l16;
            #pragma unroll
            for (int t = 0; t < 8; ++t) prow[t * 16] = acc[t][j];
        }
    }
}

// Deterministic finish: sum K-splits in fixed order, ReLU, dot with w2,
// fixed-tree block reduction over the 256 (half,h) pairs.
__global__ __launch_bounds__(256)
void nnue_reduce(const float* __restrict__ partial, const float* __restrict__ w2,
                 float* __restrict__ out, int KS)
{
    __shared__ float red[256];
    const int b    = blockIdx.x;
    const int t    = threadIdx.x;
    const int half = t >> 7;
    const int h    = t & 127;

    float s = 0.0f;
    for (int ks = 0; ks < KS; ++ks)
        s += partial[((size_t)(half * KS + ks) * BATCH + b) * HID + h];
    s = s > 0.0f ? s : 0.0f;
    red[t] = s * w2[half * HID + h];
    __syncthreads();
    #pragma unroll
    for (int off = 128; off > 0; off >>= 1) {
        if (t < off) red[t] += red[t + off];
        __syncthreads();
    }
    if (t == 0) out[b] = red[0];
}

extern "C" void kernel_launch(void* const* d_in, const int* in_sizes, int n_in,
                              void* d_out, int out_size, void* d_ws, size_t ws_size,
                              hipStream_t stream)
{
    const float* x1     = (const float*)d_in[0];
    const float* x2     = (const float*)d_in[1];
    /* d_in[2] = turn (int32): unused by the reference computation */
    const float* w_us   = (const float*)d_in[3];
    const float* w_them = (const float*)d_in[4];
    const float* w2     = (const float*)d_in[5];
    float* out = (float*)d_out;
    float* ws  = (float*)d_ws;

    const size_t bytes_per_split = (size_t)2 * BATCH * HID * sizeof(float); // 4 MiB

    int KS = 0;
    if      (ws_size >= 16 * bytes_per_split) KS = 16;
    else if (ws_size >=  8 * bytes_per_split) KS = 8;
    else if (ws_size >=  4 * bytes_per_split) KS = 4;
    else if (ws_size >=  2 * bytes_per_split) KS = 2;

    if (KS > 0) {
        dim3 grid(BATCH / 128, KS, 2);           // up to 32*16*2 blocks, 8 waves each
        nnue_gemm<false><<<grid, 256, 0, stream>>>(x1, x2, w_us, w_them, w2,
                                                   out, ws, KS, K_IN / KS);
        nnue_reduce<<<BATCH, 256, 0, stream>>>(ws, w2, out, KS);
    } else {
        nnue_zero_out<<<(BATCH + 255) / 256, 256, 0, stream>>>(out);
        dim3 grid(BATCH / 128, 1, 2);
        nnue_gemm<true><<<grid, 256, 0, stream>>>(x1, x2, w_us, w_them, w2,
                                                  out, nullptr, 1, K_IN);
    }
}